// ProjectPointNet_51153060495764
// MI455X (gfx1250) — compile-verified
//
#include <hip/hip_runtime.h>
#include <hip/hip_bf16.h>

typedef __attribute__((ext_vector_type(2))) float v2f;
typedef __attribute__((ext_vector_type(8))) float v8f;

// ---------------- problem constants ----------------
constexpr int   Bx  = 2;
constexpr int   Hh  = 64;
constexpr int   Ww  = 2048;
constexpr int   OH  = 32;
constexpr int   OW  = 1024;
constexpr int   S   = OH * OW;          // 32768
constexpr int   NSm = 16;
constexpr int   Cf  = 64;
constexpr size_t NR = (size_t)Bx * S * NSm;   // 1,048,576 rows
constexpr int   CINP0 = 68;                    // 67 padded to mult of 4

// output element offsets (floats), concatenated in reference return order
constexpr size_t O2 = (size_t)Bx * S * 3;            // new_xyz_proj
constexpr size_t O3 = O2 * 2;                        // new_points
constexpr size_t O4 = O3 + (size_t)Bx * S * 128;     // grouped_xyz

// workspace byte offsets
constexpr size_t OFF_IDX = 0;
constexpr size_t OFF_X0  = OFF_IDX + NR * sizeof(int);
constexpr size_t OFF_Y0  = OFF_X0 + NR * (size_t)CINP0 * sizeof(float);
constexpr size_t OFF_GMX = OFF_Y0 + NR * 64 * sizeof(float);
constexpr size_t OFF_GMN = OFF_GMX + (size_t)Bx * S * 128 * sizeof(float);
constexpr size_t OFF_ST  = OFF_GMN + (size_t)Bx * S * 128 * sizeof(float);
constexpr size_t OFF_XF  = OFF_ST + 512 * sizeof(float);

// ---------------- utility ----------------
__global__ void zero_kernel(float* p, int n) {
    int i = blockIdx.x * 256 + threadIdx.x;
    if (i < n) p[i] = 0.f;
}

__device__ __forceinline__ unsigned lds_addr_u32(const void* p) {
    // low 32 bits of a flat LDS pointer are the LDS byte offset (aperture rules)
    return (unsigned)(uintptr_t)p;
}

__device__ __forceinline__ void async_copy_b128(unsigned lds_byte, unsigned g_off,
                                                unsigned long long g_base) {
    asm volatile("global_load_async_to_lds_b128 %0, %1, %2"
                 :: "v"(lds_byte), "v"(g_off), "s"(g_base) : "memory");
}

// ---------------- KNN neighbor selection (per center point) ----------------
__global__ void knn_kernel(const float* __restrict__ xyzr,
                           const float* __restrict__ xyzp,
                           int* __restrict__ idxbuf,
                           float* __restrict__ out) {
    int t = blockIdx.x * 256 + threadIdx.x;           // 0 .. B*S-1
    if (t >= Bx * S) return;
    int b = t / S, s = t % S;
    int oh = s / OW, ow = s % OW;
    int hh = oh * 2, ww = ow * 2;
    int center = hh * Ww + ww;
    const float* base = xyzr + (size_t)b * Hh * Ww * 3;
    float cx = base[(size_t)center * 3 + 0];
    float cy = base[(size_t)center * 3 + 1];
    float cz = base[(size_t)center * 3 + 2];

    int   cand[25];
    float score[25];
#pragma unroll
    for (int j = 0; j < 25; ++j) {
        int dh = j / 5 - 2, dw = j % 5 - 2;
        int hn = min(max(hh + dh, 0), Hh - 1);
        int wn = min(max(ww + dw, 0), Ww - 1);
        int ci = hn * Ww + wn;
        cand[j] = ci;
        float dx = base[(size_t)ci * 3 + 0] - cx;
        float dy = base[(size_t)ci * 3 + 1] - cy;
        float dz = base[(size_t)ci * 3 + 2] - cz;
        float d2 = dx * dx + dy * dy + dz * dz;
        score[j] = (d2 < 9.0f) ? -d2 : -1e9f;
    }
    // stable top-16 (ties -> lower index first), invalid -> center (matches jax top_k + where)
    unsigned taken = 0;
    for (int n = 0; n < NSm; ++n) {
        float best = -3.0e38f;
        int   bc   = center;
        unsigned bm = 0;
#pragma unroll
        for (int j = 0; j < 25; ++j) {
            bool free = ((taken >> j) & 1u) == 0u;
            if (free && score[j] > best) { best = score[j]; bc = cand[j]; bm = 1u << j; }
        }
        taken |= bm;
        idxbuf[(size_t)t * NSm + n] = (best <= -5e8f) ? center : bc;
    }
    // outputs 1 & 2: gathered centers
    const float* basep = xyzp + (size_t)b * Hh * Ww * 3;
#pragma unroll
    for (int c = 0; c < 3; ++c) {
        out[(size_t)t * 3 + c]      = base[(size_t)center * 3 + c];
        out[O2 + (size_t)t * 3 + c] = basep[(size_t)center * 3 + c];
    }
}

// ---------------- gather + concat -> x0 [NR, 68]; emits grouped_xyz ----------------
__global__ void build_kernel(const float* __restrict__ xyzp,
                             const float* __restrict__ feat,
                             const int* __restrict__ idxbuf,
                             float* __restrict__ x0,
                             float* __restrict__ out) {
    int tid = threadIdx.x;
    size_t r = (size_t)blockIdx.x * 4 + (tid >> 6);   // row (b,s,n)
    int c = tid & 63;
    int gidx = idxbuf[r];
    int b = (int)(r / ((size_t)S * NSm));
    int s = (int)((r / NSm) % S);
    int center = (s / OW) * 2 * Ww + (s % OW) * 2;
    const float* fb = feat + ((size_t)b * Hh * Ww + (size_t)gidx) * Cf;
    float* xr = x0 + r * CINP0;
    xr[3 + c] = fb[c];
    if (c < 3) {
        const float* pb = xyzp + (size_t)b * Hh * Ww * 3;
        float gx  = pb[(size_t)gidx * 3 + c];
        float cxv = pb[(size_t)center * 3 + c];
        xr[c] = gx - cxv;
        out[O4 + r * 3 + c] = gx;
    }
    if (c == 63) xr[67] = 0.f;
}

// ---------------- WMMA f32 GEMM: Y = relu?(X*a+c) @ W + bias, with batch stats ----
// A-tile and W panel staged into LDS via GLOBAL_LOAD_ASYNC_TO_LDS_B128 (ASYNCcnt).
// Each wave owns MT 16-row M-tiles so one LDS B-fragment read feeds MT WMMAs.
// MODE 0: store Y [NR,COUT].  MODE 1: store per-16-row-group max/min (NS==WMMA M tile).
template <int CINP, int COUT, int KREAL, int ROWS, int MT, bool XFORM, int MODE>
__global__ __launch_bounds__((ROWS / (16 * MT)) * 32) void gemm_kernel(
    const float* __restrict__ X, const float* __restrict__ Wg,
    const float* __restrict__ bias, const float* __restrict__ xf,
    float* __restrict__ stats, float* __restrict__ Y,
    float* __restrict__ gmax, float* __restrict__ gmin) {
    constexpr int NW      = ROWS / (16 * MT);
    constexpr int THREADS = NW * 32;
    // pad LDS A row stride when CINP is a multiple of 64 banks (avoid 16-way conflicts)
    constexpr int ASTRIDE = CINP + (((CINP & 63) == 0) ? 2 : 0);

    __shared__ float sW[CINP * COUT];
    __shared__ float sA[ROWS * ASTRIDE];
    __shared__ float sXF[XFORM ? 2 * CINP : 1];
    __shared__ float sSum[COUT];
    __shared__ float sSq[COUT];

    const int tid = threadIdx.x;

    // ---- async stage W panel (contiguous KREAL*COUT floats) into LDS ----
    {
        unsigned long long wg = (unsigned long long)Wg;
        unsigned ldsW = lds_addr_u32(&sW[0]);
        constexpr int WCH = (KREAL * COUT) / 4;         // 16B chunks
        for (int c = tid; c < WCH; c += THREADS)
            async_copy_b128(ldsW + (unsigned)c * 16u, (unsigned)c * 16u, wg);
        // zero-fill padded K rows (k >= KREAL), disjoint from async target region
        for (int i = KREAL * COUT + tid; i < CINP * COUT; i += THREADS) sW[i] = 0.f;
    }
    // ---- async stage A tile (ROWS consecutive rows, contiguous in global) ----
    {
        const float* Xt = X + (size_t)blockIdx.x * ROWS * CINP;
        unsigned long long xg = (unsigned long long)Xt;
        unsigned ldsA = lds_addr_u32(&sA[0]);
        constexpr int CH_ROW = CINP / 4;                // 16B chunks per row
        constexpr int NCH    = ROWS * CH_ROW;
        for (int c = tid; c < NCH; c += THREADS) {
            int row = c / CH_ROW, k4 = c % CH_ROW;
            unsigned g = (unsigned)((row * CINP + k4 * 4) * 4);
            unsigned l = ldsA + (unsigned)((row * ASTRIDE + k4 * 4) * 4);
            async_copy_b128(l, g, xg);
        }
    }
    if (XFORM)
        for (int i = tid; i < 2 * CINP; i += THREADS) sXF[i] = xf[i];
    for (int i = tid; i < COUT; i += THREADS) { sSum[i] = 0.f; sSq[i] = 0.f; }

    asm volatile("s_wait_asynccnt 0x0" ::: "memory");
    __syncthreads();

    const int wave = tid >> 5, lane = tid & 31, lo = lane & 15, hi = lane >> 4;
    const int rowBase0 = (blockIdx.x * NW + wave) * 16 * MT;   // global row of tile 0
    const float* arow[MT];
#pragma unroll
    for (int mt = 0; mt < MT; ++mt)
        arow[mt] = &sA[(size_t)(wave * 16 * MT + mt * 16 + lo) * ASTRIDE];

    constexpr int NT = COUT / 16;
    v8f acc[MT][NT];
    v8f zero = {0.f, 0.f, 0.f, 0.f, 0.f, 0.f, 0.f, 0.f};
#pragma unroll
    for (int mt = 0; mt < MT; ++mt)
#pragma unroll
        for (int t = 0; t < NT; ++t) acc[mt][t] = zero;

#pragma unroll 1
    for (int k0 = 0; k0 < CINP; k0 += 4) {
        const int kk = k0 + 2 * hi;                     // A 16x4 f32 fragment layout
        v2f a[MT];
#pragma unroll
        for (int mt = 0; mt < MT; ++mt) {
            float ax = arow[mt][kk], ay = arow[mt][kk + 1];
            if (XFORM) {                                // fold prev BN+ReLU into A load
                ax = fmaxf(ax * sXF[kk]     + sXF[CINP + kk],     0.f);
                ay = fmaxf(ay * sXF[kk + 1] + sXF[CINP + kk + 1], 0.f);
            }
            a[mt].x = ax; a[mt].y = ay;
        }
#pragma unroll
        for (int t = 0; t < NT; ++t) {
            v2f b;
            b.x = sW[(size_t)kk * COUT + t * 16 + lo];
            b.y = sW[(size_t)(kk + 1) * COUT + t * 16 + lo];
#pragma unroll
            for (int mt = 0; mt < MT; ++mt)             // one B read feeds MT WMMAs
                acc[mt][t] = __builtin_amdgcn_wmma_f32_16x16x4_f32(
                    false, a[mt], false, b, (short)0, acc[mt][t], false, false);
        }
    }

#pragma unroll
    for (int mt = 0; mt < MT; ++mt) {
        const int rowBase = rowBase0 + mt * 16;
#pragma unroll
        for (int t = 0; t < NT; ++t) {
            const int j = t * 16 + lo;
            const float bj = bias[j];
            float ps = 0.f, pq = 0.f;
            float mx = -3.4e38f, mn = 3.4e38f;
#pragma unroll
            for (int v = 0; v < 8; ++v) {
                float e = acc[mt][t][v] + bj;
                acc[mt][t][v] = e;
                ps += e; pq += e * e;
                if (MODE == 1) { mx = fmaxf(mx, e); mn = fminf(mn, e); }
            }
            atomicAdd(&sSum[j], ps);
            atomicAdd(&sSq[j], pq);
            if (MODE == 0) {
#pragma unroll
                for (int v = 0; v < 8; ++v)
                    Y[(size_t)(rowBase + v + 8 * hi) * COUT + j] = acc[mt][t][v];
            } else {
                float omx = __shfl_xor(mx, 16, 32);
                float omn = __shfl_xor(mn, 16, 32);
                mx = fmaxf(mx, omx);
                mn = fminf(mn, omn);
                if (hi == 0) {                            // one group == 16-row tile
                    size_t g = (size_t)(rowBase >> 4);
                    gmax[g * COUT + j] = mx;
                    gmin[g * COUT + j] = mn;
                }
            }
        }
    }
    __syncthreads();
    for (int i = tid; i < COUT; i += THREADS) {
        atomicAdd(&stats[i], sSum[i]);
        atomicAdd(&stats[COUT + i], sSq[i]);
    }
}

// ---------------- BN finalize: stats -> per-channel (a, c) ----------------
template <int COUT>
__global__ void finalize_kernel(const float* __restrict__ stats,
                                const float* __restrict__ g,
                                const float* __restrict__ bt,
                                float* __restrict__ xf) {
    int j = threadIdx.x;
    if (j >= COUT) return;
    const float invN = 1.0f / (float)NR;
    float m   = stats[j] * invN;
    float var = fmaxf(stats[COUT + j] * invN - m * m, 0.f);
    float rs  = rsqrtf(var + 1e-5f);
    float a   = rs * g[j];
    xf[j]        = a;
    xf[COUT + j] = bt[j] - m * a;
}

// ---------------- final: pick max/min by sign(a), affine + relu ----------------
__global__ void final_kernel(const float* __restrict__ gmax,
                             const float* __restrict__ gmin,
                             const float* __restrict__ xf2,
                             float* __restrict__ out3) {
    size_t i = (size_t)blockIdx.x * 256 + threadIdx.x;
    int j = (int)(i & 127);
    float a = xf2[j], c = xf2[128 + j];
    float y = (a >= 0.f) ? gmax[i] : gmin[i];
    out3[i] = fmaxf(y * a + c, 0.f);
}

// ---------------- host launcher ----------------
extern "C" void kernel_launch(void* const* d_in, const int* in_sizes, int n_in,
                              void* d_out, int out_size, void* d_ws, size_t ws_size,
                              hipStream_t stream) {
    const float* xyzr  = (const float*)d_in[0];
    const float* xyzp  = (const float*)d_in[1];
    const float* feat  = (const float*)d_in[2];
    const float* w0    = (const float*)d_in[3];
    const float* bias0 = (const float*)d_in[4];
    const float* g0    = (const float*)d_in[5];
    const float* bt0   = (const float*)d_in[6];
    const float* w1    = (const float*)d_in[7];
    const float* bias1 = (const float*)d_in[8];
    const float* g1    = (const float*)d_in[9];
    const float* bt1   = (const float*)d_in[10];
    const float* w2    = (const float*)d_in[11];
    const float* bias2 = (const float*)d_in[12];
    const float* g2    = (const float*)d_in[13];
    const float* bt2   = (const float*)d_in[14];

    char* wsb = (char*)d_ws;
    int*   idxbuf = (int*)(wsb + OFF_IDX);
    float* x0     = (float*)(wsb + OFF_X0);
    float* y0     = (float*)(wsb + OFF_Y0);
    float* y1     = (float*)(wsb + OFF_X0);     // reuse x0 region (dead after GEMM0)
    float* gmax   = (float*)(wsb + OFF_GMX);
    float* gmin   = (float*)(wsb + OFF_GMN);
    float* stats  = (float*)(wsb + OFF_ST);     // [128 | 128 | 256]
    float* xfb    = (float*)(wsb + OFF_XF);     // [128 | 128 | 256]
    float* stats0 = stats;
    float* stats1 = stats + 128;
    float* stats2 = stats + 256;
    float* xf0 = xfb;
    float* xf1 = xfb + 128;
    float* xf2 = xfb + 256;
    float* out = (float*)d_out;

    zero_kernel<<<2, 256, 0, stream>>>(stats, 512);
    knn_kernel<<<(Bx * S) / 256, 256, 0, stream>>>(xyzr, xyzp, idxbuf, out);
    build_kernel<<<(unsigned)(NR / 4), 256, 0, stream>>>(xyzp, feat, idxbuf, x0, out);

    // layer 0: 67->64, 128-row tiles, 4 waves x (2 x 16-row M-tiles)
    gemm_kernel<CINP0, 64, 67, 128, 2, false, 0><<<(unsigned)(NR / 128), 128, 0, stream>>>(
        x0, w0, bias0, nullptr, stats0, y0, nullptr, nullptr);
    finalize_kernel<64><<<1, 64, 0, stream>>>(stats0, g0, bt0, xf0);

    // layer 1: 64->64 with BN+ReLU folded into A load
    gemm_kernel<64, 64, 64, 128, 2, true, 0><<<(unsigned)(NR / 128), 128, 0, stream>>>(
        y0, w1, bias1, xf0, stats1, y1, nullptr, nullptr);
    finalize_kernel<64><<<1, 64, 0, stream>>>(stats1, g1, bt1, xf1);

    // layer 2: 64->128, emits per-neighborhood max/min instead of 512MB activations
    gemm_kernel<64, 128, 64, 64, 2, true, 1><<<(unsigned)(NR / 64), 64, 0, stream>>>(
        y1, w2, bias2, xf1, stats2, nullptr, gmax, gmin);
    finalize_kernel<128><<<1, 128, 0, stream>>>(stats2, g2, bt2, xf2);

    final_kernel<<<(unsigned)(((size_t)Bx * S * 128) / 256), 256, 0, stream>>>(
        gmax, gmin, xf2, out + O3);
}